// RefinementLoss_3307124818604
// MI455X (gfx1250) — compile-verified
//
#include <hip/hip_runtime.h>
#include <hip/hip_bf16.h>
#include <math.h>

typedef float v2f __attribute__((ext_vector_type(2)));
typedef float v8f __attribute__((ext_vector_type(8)));

#define BATCH      8
#define NPTS       4096
#define DIM        512
#define KNN        8
#define SELF_MASK_C 1000000.0f
#define CHUNK      2048   // column points staged in LDS per pass (32KB)

// ---------------------------------------------------------------------------
// ws layout (floats): [0]=sum rowmin d(P,P0)  [1]=sum rowmin d(P0,P)
//                     [2]=smooth sq-diff sum  [3]=sum of (-cos) over batches
// ---------------------------------------------------------------------------

__global__ void zero_ws_kernel(float* ws) {
    if (threadIdx.x < 8) ws[threadIdx.x] = 0.0f;
}

// ---------------------------------------------------------------------------
// Chamfer one-direction: for every row p of Apts, min over all q of Bpts of
// |p-q|^2, summed into *accum.  d = |p|^2 + (-2 p.q + |q|^2), the parenthesized
// term computed by V_WMMA_F32_16X16X4_F32 with A=(x,y,z,1), B=(-2x,-2y,-2z,|q|^2).
// Block: 256 thr = 8 waves, each wave owns 16 rows.  Grid: 8 batches * 32.
// ---------------------------------------------------------------------------
__global__ __launch_bounds__(256) void chamfer_rowmin_kernel(
        const float* __restrict__ Apts, const float* __restrict__ Bpts,
        float* __restrict__ accum) {
    __shared__ float Q[CHUNK * 4];              // (-2x,-2y,-2z,|q|^2) per point

    const int b    = blockIdx.x >> 5;           // batch
    const int rb   = blockIdx.x & 31;           // row block (128 rows)
    const int lane = threadIdx.x & 31;
    const int wave = threadIdx.x >> 5;

    const float* Ab = Apts + (size_t)b * NPTS * 3;
    const float* Bb = Bpts + (size_t)b * NPTS * 3;

    // --- A operand: 16x4 f32, lanes 0-15 hold (K0,K1)=(x,y), lanes 16-31 (K2,K3)=(z,1)
    const int row = rb * 128 + wave * 16 + (lane & 15);
    const float px = Ab[row * 3 + 0];
    const float py = Ab[row * 3 + 1];
    const float pz = Ab[row * 3 + 2];
    const float pn = px * px + py * py + pz * pz;   // |p|^2 for this lane's row
    v2f a;
    if (lane < 16) { a.x = px; a.y = py; }
    else           { a.x = pz; a.y = 1.0f; }

    v8f acc;                                    // running min of (-2p.q+|q|^2)
    #pragma unroll
    for (int i = 0; i < 8; ++i) acc[i] = 3.0e38f;

    const int comp = (lane < 16) ? 0 : 2;       // B: lanes 0-15 rows K0,K1; 16-31 rows K2,K3
    const int nsub = lane & 15;

    for (int cb = 0; cb < NPTS; cb += CHUNK) {
        // cooperative fill + transform of the column chunk
        #pragma unroll
        for (int i = 0; i < CHUNK / 256; ++i) {
            const int n = threadIdx.x + i * 256;
            const int g = cb + n;
            const float qx = Bb[g * 3 + 0];
            const float qy = Bb[g * 3 + 1];
            const float qz = Bb[g * 3 + 2];
            float4 qv = make_float4(-2.0f * qx, -2.0f * qy, -2.0f * qz,
                                    qx * qx + qy * qy + qz * qz);
            *reinterpret_cast<float4*>(&Q[n * 4]) = qv;
        }
        __syncthreads();

        for (int t = 0; t < CHUNK / 16; ++t) {
            v2f bop = *reinterpret_cast<const v2f*>(&Q[(t * 16 + nsub) * 4 + comp]);
            v8f c = {};
            // 8 args: (neg_a, A, neg_b, B, c_mod, C, reuse_a, reuse_b)
            c = __builtin_amdgcn_wmma_f32_16x16x4_f32(
                    false, a, false, bop, (short)0, c, false, false);
            #pragma unroll
            for (int i = 0; i < 8; ++i) acc[i] = fminf(acc[i], c[i]);
        }
        __syncthreads();
    }

    // min across the 16 lanes of each half (D layout: n = lane%16, m = vgpr + 8*(lane>=16))
    #pragma unroll
    for (int off = 1; off < 16; off <<= 1) {
        #pragma unroll
        for (int i = 0; i < 8; ++i) acc[i] = fminf(acc[i], __shfl_xor(acc[i], off));
    }

    // add |p_m|^2 per row and sum this half's 8 rows
    const int hb = (lane >= 16) ? 8 : 0;
    float s = 0.0f;
    #pragma unroll
    for (int i = 0; i < 8; ++i) {
        s += acc[i] + __shfl(pn, i + hb);
    }
    if (lane == 0 || lane == 16) atomicAdd(accum, s);
}

// ---------------------------------------------------------------------------
// Smoothness: per-thread kNN (k=8) with the whole batch cloud in LDS (48KB of
// the WGP's 320KB).  Sorted-insert keeps the 8 nearest fully in registers.
// Grid: 8 batches * 16 segments; block 256.
// ---------------------------------------------------------------------------
__global__ __launch_bounds__(256) void smooth_knn_kernel(
        const float* __restrict__ P, float* __restrict__ accum) {
    __shared__ float sx[NPTS], sy[NPTS], sz[NPTS];
    const int b   = blockIdx.x >> 4;
    const int seg = blockIdx.x & 15;
    const float* Pb = P + (size_t)b * NPTS * 3;

    #pragma unroll
    for (int k = 0; k < NPTS / 256; ++k) {
        const int j = threadIdx.x + k * 256;
        sx[j] = Pb[j * 3 + 0];
        sy[j] = Pb[j * 3 + 1];
        sz[j] = Pb[j * 3 + 2];
    }
    __syncthreads();

    const int i = seg * 256 + threadIdx.x;
    const float px = sx[i], py = sy[i], pz = sz[i];

    float nd[KNN], nx[KNN], ny[KNN], nz[KNN];
    #pragma unroll
    for (int s = 0; s < KNN; ++s) { nd[s] = 3.0e38f; nx[s] = 0.f; ny[s] = 0.f; nz[s] = 0.f; }

    for (int j = 0; j < NPTS; ++j) {
        const float qx = sx[j], qy = sy[j], qz = sz[j];
        const float dx = px - qx, dy = py - qy, dz = pz - qz;
        float d = dx * dx + dy * dy + dz * dz;
        if (j == i) d += SELF_MASK_C;           // reference masks self with +1e6
        if (d < nd[KNN - 1]) {                  // rare after warm-up
            float cd = d, cx = qx, cy = qy, cz = qz;
            #pragma unroll
            for (int s = 0; s < KNN; ++s) {     // unrolled sorted insert (ascending)
                const bool c = cd < nd[s];
                const float td = nd[s], tx = nx[s], ty = ny[s], tz = nz[s];
                nd[s] = c ? cd : td; nx[s] = c ? cx : tx;
                ny[s] = c ? cy : ty; nz[s] = c ? cz : tz;
                cd = c ? td : cd; cx = c ? tx : cx;
                cy = c ? ty : cy; cz = c ? tz : cz;
            }
        }
    }

    float mx = 0.f, my = 0.f, mz = 0.f;
    #pragma unroll
    for (int s = 0; s < KNN; ++s) { mx += nx[s]; my += ny[s]; mz += nz[s]; }
    mx *= 0.125f; my *= 0.125f; mz *= 0.125f;
    const float dx = px - mx, dy = py - my, dz = pz - mz;
    float c = dx * dx + dy * dy + dz * dz;

    #pragma unroll
    for (int off = 1; off < 32; off <<= 1) c += __shfl_xor(c, off);
    if ((threadIdx.x & 31) == 0) atomicAdd(accum, c);
}

// ---------------------------------------------------------------------------
// Text cosine similarity: one wave per batch row of (8, 512).
// ---------------------------------------------------------------------------
__global__ void text_sim_kernel(const float* __restrict__ z3d,
                                const float* __restrict__ zt,
                                float* __restrict__ accum) {
    const int wave = threadIdx.x >> 5;
    const int lane = threadIdx.x & 31;
    const float* a0 = z3d + wave * DIM;
    const float* b0 = zt  + wave * DIM;
    float aa = 0.f, bb = 0.f, ab = 0.f;
    for (int i = lane; i < DIM; i += 32) {
        const float av = a0[i], bv = b0[i];
        aa += av * av; bb += bv * bv; ab += av * bv;
    }
    #pragma unroll
    for (int off = 1; off < 32; off <<= 1) {
        aa += __shfl_xor(aa, off);
        bb += __shfl_xor(bb, off);
        ab += __shfl_xor(ab, off);
    }
    if (lane == 0) atomicAdd(accum, -(ab / (sqrtf(aa) * sqrtf(bb))));
}

// ---------------------------------------------------------------------------
// Combine partial sums -> (L_total, L_text, L_stick, L_smooth)
// ---------------------------------------------------------------------------
__global__ void finalize_kernel(const float* __restrict__ ws, float* __restrict__ out) {
    if (threadIdx.x == 0 && blockIdx.x == 0) {
        const float invBN = 1.0f / (8.0f * 4096.0f);
        const float L_text   = ws[3] / 8.0f;
        const float L_stick  = (ws[0] + ws[1]) * invBN;
        const float L_smooth = ws[2] * (invBN / 3.0f);
        out[0] = 0.5f * L_text + 2.0f * L_stick + 0.1f * L_smooth;
        out[1] = L_text;
        out[2] = L_stick;
        out[3] = L_smooth;
    }
}

extern "C" void kernel_launch(void* const* d_in, const int* in_sizes, int n_in,
                              void* d_out, int out_size, void* d_ws, size_t ws_size,
                              hipStream_t stream) {
    const float* P   = (const float*)d_in[0];   // (8, 4096, 3)
    const float* P0  = (const float*)d_in[1];   // (8, 4096, 3)
    const float* z3d = (const float*)d_in[2];   // (8, 512)
    const float* zt  = (const float*)d_in[3];   // (8, 512)
    float* out = (float*)d_out;                 // 4 floats
    float* ws  = (float*)d_ws;

    zero_ws_kernel<<<1, 32, 0, stream>>>(ws);
    text_sim_kernel<<<1, 256, 0, stream>>>(z3d, zt, ws + 3);
    chamfer_rowmin_kernel<<<BATCH * 32, 256, 0, stream>>>(P,  P0, ws + 0); // P -> P0
    chamfer_rowmin_kernel<<<BATCH * 32, 256, 0, stream>>>(P0, P,  ws + 1); // P0 -> P
    smooth_knn_kernel<<<BATCH * 16, 256, 0, stream>>>(P, ws + 2);
    finalize_kernel<<<1, 64, 0, stream>>>(ws, out);
}